// _CompressJPEG_21311627723261
// MI455X (gfx1250) — compile-verified
//
#include <hip/hip_runtime.h>
#include <hip/hip_bf16.h>
#include <math.h>

typedef float v2f __attribute__((ext_vector_type(2)));
typedef float v8f __attribute__((ext_vector_type(8)));

#define PI_OVER_16 0.19634954084936207f
#define INV_SQRT2  0.70710678118654752f

__device__ __constant__ float c_qy[64] = {
  16,11,10,16,24,40,51,61,
  12,12,14,19,26,58,60,55,
  14,13,16,24,40,57,69,56,
  14,17,22,29,51,87,80,62,
  18,22,37,56,68,109,103,77,
  24,35,55,64,81,104,113,92,
  49,64,78,87,103,121,120,101,
  72,92,95,98,112,100,103,99 };

__device__ __constant__ float c_qc[64] = {
  17,18,24,47,99,99,99,99,
  18,21,26,66,99,99,99,99,
  24,26,56,99,99,99,99,99,
  47,66,99,99,99,99,99,99,
  99,99,99,99,99,99,99,99,
  99,99,99,99,99,99,99,99,
  99,99,99,99,99,99,99,99,
  99,99,99,99,99,99,99,99 };

// ---------------------------------------------------------------------------
// Kernel 1: RGB -> YCbCr (x255 + shift) with fused 2x2 average pooling of Cb/Cr.
// One thread per 2x2 pixel quad. Fully coalesced float2 loads/stores.
// ---------------------------------------------------------------------------
__global__ void __launch_bounds__(256) ycc_pool_kernel(
    const float* __restrict__ x,   // (16,3,512,512)
    float* __restrict__ Y,         // (16,512,512)
    float* __restrict__ Cb,        // (16,256,256)
    float* __restrict__ Cr)        // (16,256,256)
{
  int idx = blockIdx.x * 256 + threadIdx.x;     // quad id, 16*256*256 total
  int b   = idx >> 16;
  int rem = idx & 65535;
  int qy  = rem >> 8;
  int qx  = rem & 255;

  const size_t plane = (size_t)512 * 512;
  const float* xb = x + (size_t)b * 3 * plane + (size_t)(2 * qy) * 512 + 2 * qx;

  float2 r0 = *(const float2*)(xb);
  float2 r1 = *(const float2*)(xb + 512);
  float2 g0 = *(const float2*)(xb + plane);
  float2 g1 = *(const float2*)(xb + plane + 512);
  float2 b0 = *(const float2*)(xb + 2 * plane);
  float2 b1 = *(const float2*)(xb + 2 * plane + 512);

  float yv[4], cbs = 0.f, crs = 0.f;
  float R[4] = { r0.x, r0.y, r1.x, r1.y };
  float G[4] = { g0.x, g0.y, g1.x, g1.y };
  float B[4] = { b0.x, b0.y, b1.x, b1.y };
#pragma unroll
  for (int i = 0; i < 4; ++i) {
    float Rf = R[i] * 255.f, Gf = G[i] * 255.f, Bf = B[i] * 255.f;
    yv[i] = 0.299f * Rf + 0.587f * Gf + 0.114f * Bf;
    cbs  += -0.168736f * Rf - 0.331264f * Gf + 0.5f * Bf + 128.f;
    crs  += 0.5f * Rf - 0.418688f * Gf - 0.081312f * Bf + 128.f;
  }

  float* Yb = Y + (size_t)b * plane + (size_t)(2 * qy) * 512 + 2 * qx;
  *(float2*)(Yb)        = make_float2(yv[0], yv[1]);
  *(float2*)(Yb + 512)  = make_float2(yv[2], yv[3]);
  size_t ci = (size_t)b * 256 * 256 + (size_t)qy * 256 + qx;
  Cb[ci] = cbs * 0.25f;
  Cr[ci] = crs * 0.25f;
}

// ---------------------------------------------------------------------------
// Kernel 2: per-wave 16x16 tile = four 8x8 blocks. Two chained f32 WMMA
// products with a block-diagonal cosine basis compute all four 8x8 DCTs:
//   T   = Mt * X^T      (B operand = X loaded in A-layout pattern)
//   OUT = Mt * T^T      (B operand = A-layout of T via LDS round-trip)
// Then scale / quantize / diff-round and store into block-split layout.
// ---------------------------------------------------------------------------
__global__ void __launch_bounds__(256) dct_quant_kernel(
    const float* __restrict__ plane,   // (16, S, S)
    float* __restrict__ out,           // (16, NB, 8, 8)
    const float* __restrict__ factor,  // (16,)
    int S, int tilesPerRow, int isY)
{
  __shared__ float lds[8][256];

  int lane = threadIdx.x & 31;
  int w    = threadIdx.x >> 5;
  int lm   = lane & 15;
  int hi   = lane >> 4;     // half-wave select

  int tilesPerImg = tilesPerRow * tilesPerRow;
  int tile = blockIdx.x * 8 + w;
  int b  = tile / tilesPerImg;
  int r2 = tile - b * tilesPerImg;
  int ty = r2 / tilesPerRow;
  int tx = r2 - ty * tilesPerRow;

  // Constant basis Mt in WMMA A-layout. Mt[p,q] = blockdiag cos((2(q%8)+1)(p%8)pi/16)
  v2f mt[4];
#pragma unroll
  for (int k = 0; k < 4; ++k) {
    int q0 = 4 * k + 2 * hi;
    int p  = lm;
    float m0 = ((p < 8) == (q0 < 8))
        ? __cosf((2.f * (float)(q0 & 7) + 1.f) * (float)(p & 7) * PI_OVER_16) : 0.f;
    int q1 = q0 + 1;
    float m1 = ((p < 8) == (q1 < 8))
        ? __cosf((2.f * (float)(q1 & 7) + 1.f) * (float)(p & 7) * PI_OVER_16) : 0.f;
    mt[k].x = m0; mt[k].y = m1;
  }

  // Load X (minus 128) in A-layout pattern == B-layout of X^T.
  const float* tp = plane + ((size_t)b * S + (size_t)(ty * 16 + lm)) * S + tx * 16;
  v2f xa[4];
#pragma unroll
  for (int k = 0; k < 4; ++k) {
    float2 t = *(const float2*)(tp + 4 * k + 2 * hi);
    xa[k].x = t.x - 128.f;
    xa[k].y = t.y - 128.f;
  }

  // Stage 1: T = Mt * X^T  (four chained 16x16x4 f32 WMMAs)
  v8f acc = {0.f,0.f,0.f,0.f,0.f,0.f,0.f,0.f};
#pragma unroll
  for (int k = 0; k < 4; ++k)
    acc = __builtin_amdgcn_wmma_f32_16x16x4_f32(false, mt[k], false, xa[k],
                                                (short)0, acc, false, false);

  // D-layout -> LDS (row-major 16x16), then reload rows as A-layout of T.
#pragma unroll
  for (int r = 0; r < 8; ++r)
    lds[w][(r + 8 * hi) * 16 + lm] = acc[r];

  asm volatile("s_wait_dscnt 0" ::: "memory");   // same-wave LDS RAW ordering

  v2f ta[4];
#pragma unroll
  for (int k = 0; k < 4; ++k) {
    float2 t = *(const float2*)&lds[w][lm * 16 + 4 * k + 2 * hi];
    ta[k].x = t.x; ta[k].y = t.y;
  }

  // Stage 2: OUT = Mt * T^T = Mt * X * M16
  v8f o = {0.f,0.f,0.f,0.f,0.f,0.f,0.f,0.f};
#pragma unroll
  for (int k = 0; k < 4; ++k)
    o = __builtin_amdgcn_wmma_f32_16x16x4_f32(false, mt[k], false, ta[k],
                                              (short)0, o, false, false);

  // Quantize + diff-round + store. Lane l, reg r holds OUT[r+8*hi, lm]:
  //   u = r, v = lm%8, block (bi=hi, bj=lm/8).
  float rfac = 1.0f / factor[b];
  const float* qt = isY ? c_qy : c_qc;
  int v  = lm & 7;
  int bj = lm >> 3;
  int blocksPerRow = tilesPerRow * 2;
  int nb = (ty * 2 + hi) * blocksPerRow + tx * 2 + bj;
  size_t NB = (size_t)tilesPerImg * 4;
  float av = (v == 0) ? INV_SQRT2 : 1.f;
  float* ob = out + ((size_t)b * NB + (size_t)nb) * 64 + v;
#pragma unroll
  for (int r = 0; r < 8; ++r) {
    float au    = (r == 0) ? INV_SQRT2 : 1.f;
    float scale = au * av * 0.25f;
    float val   = o[r] * scale * rfac / qt[r * 8 + v];
    float rd    = rintf(val);                 // round-half-even == jnp.round
    float d     = val - rd;
    ob[(size_t)r * 8] = rd + d * d * d;
  }
}

extern "C" void kernel_launch(void* const* d_in, const int* in_sizes, int n_in,
                              void* d_out, int out_size, void* d_ws, size_t ws_size,
                              hipStream_t stream) {
  const float* x      = (const float*)d_in[0];
  const float* factor = (const float*)d_in[1];

  float* Y  = (float*)d_ws;                       // 16*512*512
  float* Cb = Y  + (size_t)16 * 512 * 512;        // 16*256*256
  float* Cr = Cb + (size_t)16 * 256 * 256;        // 16*256*256

  float* yout  = (float*)d_out;                   // 16*4096*64
  float* cbout = yout  + (size_t)16 * 4096 * 64;  // 16*1024*64
  float* crout = cbout + (size_t)16 * 1024 * 64;  // 16*1024*64

  // 16*256*256 quads / 256 threads = 4096 blocks
  ycc_pool_kernel<<<4096, 256, 0, stream>>>(x, Y, Cb, Cr);

  // Y: 16 * (512/16)^2 = 16384 tiles, 8 waves/block -> 2048 blocks
  dct_quant_kernel<<<2048, 256, 0, stream>>>(Y,  yout,  factor, 512, 32, 1);
  // Chroma: 16 * (256/16)^2 = 4096 tiles -> 512 blocks each
  dct_quant_kernel<<<512, 256, 0, stream>>>(Cb, cbout, factor, 256, 16, 0);
  dct_quant_kernel<<<512, 256, 0, stream>>>(Cr, crout, factor, 256, 16, 0);
}